// GridQueryAndGroup_63625645523258
// MI455X (gfx1250) — compile-verified
//
#include <hip/hip_runtime.h>

// ---------------------------------------------------------------------------
// GridQueryAndGroup for MI455X (gfx1250, wave32)
//   phase 1: pack (x,y,z,|x|^2) per point                      [prep_pack]
//   phase 2: ball query on the matrix pipe: 2x V_WMMA_F32_16X16X4_F32 per
//            iteration over 32 coalesced points; wave32 ballots + uniform
//            zero-hit skip + per-row ascending first-32 select [ball_query_wmma]
//   phase 3: vectorized gather/concat, b128 loads+stores       [group_gather4]
// ---------------------------------------------------------------------------

typedef __attribute__((ext_vector_type(2))) float v2f;
typedef __attribute__((ext_vector_type(8))) float v8f;

#define N_PTS   20000
#define BDIM    2
#define PDIM    2048
#define BP      (BDIM * PDIM)     // 4096 query points total
#define CFEAT   64
#define NSAMP   32
#define RADIUS2 0.0064f           // 0.08^2

// --------------------------------------------------------------------------
// Pack xyz into float4 (x, y, z, |x|^2): one b128 load per 1-4 points in the
// scan loop instead of three scalar loads + recomputed norms.
// --------------------------------------------------------------------------
__global__ void prep_pack(const float* __restrict__ xyz,
                          float4* __restrict__ packed, int n) {
  int i = blockIdx.x * blockDim.x + threadIdx.x;
  if (i < n) {
    float x = xyz[3 * i + 0];
    float y = xyz[3 * i + 1];
    float z = xyz[3 * i + 2];
    packed[i] = make_float4(x, y, z, x * x + y * y + z * z);
  }
}

// --------------------------------------------------------------------------
// One wave owns 16 query rows. Per iteration it consumes 32 consecutive
// points with ZERO redundant VMEM (lane l loads point n0+l), builds two
// B operands (4x16) via 4 DS-permutes, and fires two WMMAs:
//   A (16x4): row m = (-2qx, -2qy, -2qz, 1)
//   B (4x16): col n = ( xx,  xy,  xz, |x|^2)
//   C        : c[v] = |q|^2 broadcast per row
//   => D = |q|^2 + |x|^2 - 2 q.x  (full f32 d2, identical math to reference)
// Ballot per D-VGPR: bits[15:0] = row v, bits[31:16] = row v+8.
// Uniform skip when the whole 16x32 tile has no hits (~33% of tiles).
// --------------------------------------------------------------------------
__global__ void __launch_bounds__(32) ball_query_wmma(
    const float4* __restrict__ packed, const float* __restrict__ new_xyz,
    int* __restrict__ idx_out, int* __restrict__ idn_out) {
  const int  lane = threadIdx.x;        // 0..31
  const int  rl   = lane & 15;          // row / col id within tile
  const bool hi   = lane >= 16;
  const int  qbase = blockIdx.x * 16;   // flat (b*P+p) of tile row 0

  // Each lane loads the query for row (lane&15); lanes 16-31 mirror 0-15.
  const int q = qbase + rl;
  const float qx = new_xyz[3 * q + 0];
  const float qy = new_xyz[3 * q + 1];
  const float qz = new_xyz[3 * q + 2];
  const float q2 = qx * qx + qy * qy + qz * qz;

  // A operand (16x4 f32: V0 = K0|K2, V1 = K1|K3 across half-waves)
  v2f a;
  a.x = hi ? (-2.0f * qz) : (-2.0f * qx);
  a.y = hi ? 1.0f         : (-2.0f * qy);

  // C operand: c[v] = |q|^2 of row (v + (hi?8:0))
  v8f c;
#pragma unroll
  for (int v = 0; v < 8; ++v)
    c[v] = __shfl(q2, v + (hi ? 8 : 0), 32);

  int  cnt = 0;
  int  firstIdx = 0;
  int* myidx = idx_out + (size_t)(qbase + rl) * NSAMP;  // owned by lanes 0-15

  for (int n0 = 0; n0 < N_PTS; n0 += 32) {   // 20000 % 32 == 0, no tail
    const float4 P = packed[n0 + lane];       // coalesced, no redundancy
    if (n0 + 128 < N_PTS)                     // 4 iterations ahead
      __builtin_prefetch(&packed[n0 + 128 + lane], 0, 1);

    // Chunk A = points n0..n0+15 (data in lanes 0-15),
    // chunk B = points n0+16..n0+31 (data in lanes 16-31).
    const float zA = __shfl(P.z, rl, 32);        // z  of point n0+rl
    const float wA = __shfl(P.w, rl, 32);        // x2 of point n0+rl
    const float xB = __shfl(P.x, rl + 16, 32);   // x  of point n0+16+rl
    const float yB = __shfl(P.y, rl + 16, 32);   // y  of point n0+16+rl
    v2f bA, bB;
    bA.x = hi ? zA  : P.x;   bA.y = hi ? wA  : P.y;
    bB.x = hi ? P.z : xB;    bB.y = hi ? P.w : yB;

    const v8f dA = __builtin_amdgcn_wmma_f32_16x16x4_f32(
        false, a, false, bA, (short)0, c, false, false);
    const v8f dB = __builtin_amdgcn_wmma_f32_16x16x4_f32(
        false, a, false, bB, (short)0, c, false, false);

    unsigned balA[8], balB[8];
#pragma unroll
    for (int v = 0; v < 8; ++v) {
      balA[v] = (unsigned)__builtin_amdgcn_ballot_w32(dA[v] < RADIUS2);
      balB[v] = (unsigned)__builtin_amdgcn_ballot_w32(dB[v] < RADIUS2);
    }

    // Uniform SALU skip: most 16x32 tiles contain zero in-ball points.
    unsigned any = 0;
#pragma unroll
    for (int v = 0; v < 8; ++v) any |= balA[v] | balB[v];
    if (any == 0u) continue;

    if (!hi && cnt < NSAMP) {
      // Select this lane's two 16-bit row masks from the uniform ballots.
      unsigned rA = 0, rB = 0;
#pragma unroll
      for (int v = 0; v < 8; ++v) {
        rA = ((lane & 7) == v) ? balA[v] : rA;
        rB = ((lane & 7) == v) ? balB[v] : rB;
      }
      const int sh = (lane >> 3) * 16;
      unsigned rowmask = ((rA >> sh) & 0xFFFFu) |
                         (((rB >> sh) & 0xFFFFu) << 16);
      // Peel ascending: bit i  ->  point n0 + i (preserves reference order).
      while (rowmask && cnt < NSAMP) {
        const int bit = __builtin_ctz(rowmask);
        rowmask &= rowmask - 1;
        const int pi = n0 + bit;
        myidx[cnt] = pi;
        if (cnt == 0) firstIdx = pi;
        ++cnt;
      }
    }

    // Uniform early exit once all 16 rows are full (~26% of scan saved).
    const unsigned done =
        (unsigned)__builtin_amdgcn_ballot_w32(hi || cnt >= NSAMP);
    if (done == 0xFFFFFFFFu) break;
  }

  if (!hi) {
    // Pad with first found index (0 if empty), reference semantics.
    const int f = (cnt > 0) ? firstIdx : 0;
    for (int s = cnt; s < NSAMP; ++s) myidx[s] = f;
    // idn: 1 for valid slots; all-ones if the ball is empty.
    int* myidn = idn_out + (size_t)(qbase + rl) * NSAMP;
#pragma unroll
    for (int s = 0; s < NSAMP; ++s)
      myidn[s] = (s < cnt || cnt == 0) ? 1 : 0;
  }
}

// --------------------------------------------------------------------------
// Gather phase: one thread per 4 consecutive samples of (B, 3+C, P, S).
// b128 idx load + b128 store; gathers are L2-resident (features 5.1MB,
// xyz 240KB, idx 512KB << 192MB L2). This phase carries the real HBM
// traffic (~35MB stores -> ~1.5us floor at 23.3 TB/s).
// --------------------------------------------------------------------------
__global__ void group_gather4(const float* __restrict__ xyz,
                              const float* __restrict__ new_xyz,
                              const float* __restrict__ features,
                              const int* __restrict__ idx,
                              float* __restrict__ out, long long total4) {
  long long t = (long long)blockIdx.x * blockDim.x + threadIdx.x;
  if (t >= total4) return;
  const int s4 = (int)(t % (NSAMP / 4)) * 4;     // sample group start
  long long r = t / (NSAMP / 4);
  const int p = (int)(r % PDIM);  r /= PDIM;
  const int cch = (int)(r % (3 + CFEAT));  r /= (3 + CFEAT);
  const int b = (int)r;
  const int bp = b * PDIM + p;

  const int4 iv = *(const int4*)&idx[(size_t)bp * NSAMP + s4];
  float4 val;
  if (cch < 3) {
    const float qv = new_xyz[3 * bp + cch];
    val.x = xyz[3 * iv.x + cch] - qv;
    val.y = xyz[3 * iv.y + cch] - qv;
    val.z = xyz[3 * iv.z + cch] - qv;
    val.w = xyz[3 * iv.w + cch] - qv;
  } else {
    const float* f = features + (size_t)(cch - 3) * N_PTS;
    val.x = f[iv.x];
    val.y = f[iv.y];
    val.z = f[iv.z];
    val.w = f[iv.w];
  }
  *(float4*)&out[t * 4] = val;
}

// --------------------------------------------------------------------------
extern "C" void kernel_launch(void* const* d_in, const int* in_sizes, int n_in,
                              void* d_out, int out_size, void* d_ws,
                              size_t ws_size, hipStream_t stream) {
  const float* xyz      = (const float*)d_in[0];  // (20000, 3)
  const float* new_xyz  = (const float*)d_in[1];  // (2, 2048, 3)
  const float* features = (const float*)d_in[2];  // (64, 20000)

  // workspace: [packed float4 x N][idx int32 x BP*32]
  float4* packed = (float4*)d_ws;
  int*    idx_ws = (int*)((char*)d_ws + (size_t)N_PTS * sizeof(float4));

  const long long feat_elems = (long long)BDIM * (3 + CFEAT) * PDIM * NSAMP;
  float* out_feat = (float*)d_out;
  int*   idn_out  = (int*)d_out + feat_elems;  // idn int32 bits in tail

  prep_pack<<<(N_PTS + 255) / 256, 256, 0, stream>>>(xyz, packed, N_PTS);
  ball_query_wmma<<<BP / 16, 32, 0, stream>>>(packed, new_xyz, idx_ws, idn_out);
  const long long total4 = feat_elems / 4;
  group_gather4<<<(int)((total4 + 255) / 256), 256, 0, stream>>>(
      xyz, new_xyz, features, idx_ws, out_feat, total4);
}